// BertBiLSTMCRF_29884382445716
// MI455X (gfx1250) — compile-verified
//
#include <hip/hip_runtime.h>
#include <stdint.h>

#define BB 1024
#define TT 512
#define KK 64
#define NEGV (-10000.0f)

typedef float v2f __attribute__((ext_vector_type(2)));
typedef float v8f __attribute__((ext_vector_type(8)));

// ---------------------------------------------------------------------------
// prep: build E = exp(transitions) laid out as WMMA B-tiles (f32 16x16x4).
// ebuf[(j*16 + kk)*32 + lane]: lane (h=lane>>4, ln=lane&15) holds
//   {E[n][p0], E[n][p0+1]} with p0 = 4*kk + 2*h, n = 16*j + ln.
// ---------------------------------------------------------------------------
__global__ void crf_prep(const float* __restrict__ trans, v2f* __restrict__ ebuf) {
    int idx = blockIdx.x * blockDim.x + threadIdx.x;
    if (idx >= 2048) return;
    int lane = idx & 31;
    int kk = (idx >> 5) & 15;
    int j = idx >> 9;
    int h = lane >> 4, ln = lane & 15;
    int p0 = 4 * kk + 2 * h;
    int n = 16 * j + ln;
    v2f e;
    e.x = __expf(trans[n * KK + p0]);
    e.y = __expf(trans[n * KK + p0 + 1]);
    ebuf[idx] = e;
}

// ---------------------------------------------------------------------------
// Forward recurrence as per-step f32 WMMA GEMM + log/exp rescaling, then
// terminal logsumexp and gold-path score -> nll.
// Block = 128 threads (4 waves); block owns 16 batch rows; wave j owns the
// 16-state N-tile j. alpha/v in LDS with 65-stride padding.
// Two independent WMMA accumulator chains (even/odd k-chunks) for in-wave
// ILP: at ~1 wave/SIMD occupancy there is no cross-wave latency hiding.
// ---------------------------------------------------------------------------
__global__ __launch_bounds__(128) void crf_fwd(
    const float* __restrict__ feats, const int* __restrict__ tags,
    const float* __restrict__ trans, const v2f* __restrict__ ebuf,
    float* __restrict__ nll)
{
    __shared__ float alphaS[16 * 65];
    __shared__ float vS[16 * 65];
    __shared__ float mS[16];
    __shared__ float pmax[16 * 8];

    const int tid = threadIdx.x;
    const int wv = tid >> 5, lane = tid & 31;
    const int h = lane >> 4, ln = lane & 15;
    const int base = blockIdx.x * 16;

    // preload this wave's 16 B-tiles of E (32 VGPRs) - stays live all T steps
    v2f eb[16];
#pragma unroll
    for (int kk = 0; kk < 16; ++kk) eb[kk] = ebuf[(wv * 16 + kk) * 32 + lane];

    // alpha0 = NEG everywhere except START(=0) column
    for (int i = tid; i < 16 * KK; i += 128) {
        int m = i >> 6, p = i & 63;
        alphaS[m * 65 + p] = (p == 0) ? 0.0f : NEGV;
    }
    __syncthreads();

    for (int t = 0; t < TT; ++t) {
        // per-row max (8 threads per row, then 16-thread final reduce)
        {
            int m = tid & 15, seg = tid >> 4;
            const float* r = &alphaS[m * 65 + seg * 8];
            float mx = -3.0e38f;
#pragma unroll
            for (int i = 0; i < 8; ++i) mx = fmaxf(mx, r[i]);
            pmax[m * 8 + seg] = mx;
        }
        __syncthreads();
        if (tid < 16) {
            float mx = -3.0e38f;
#pragma unroll
            for (int i = 0; i < 8; ++i) mx = fmaxf(mx, pmax[tid * 8 + i]);
            mS[tid] = mx;
        }
        __syncthreads();
        // v = exp(alpha - rowmax)  in (0,1]
        for (int i = tid; i < 16 * KK; i += 128) {
            int m = i >> 6, p = i & 63;
            vS[m * 65 + p] = __expf(alphaS[m * 65 + p] - mS[m]);
        }
        __syncthreads();

        // D[16x16] = V[16x64] * E^T[64x16] via 16 x wmma_f32_16x16x4,
        // split into two independent accumulator chains for ILP.
        v8f c0 = {0.f, 0.f, 0.f, 0.f, 0.f, 0.f, 0.f, 0.f};
        v8f c1 = {0.f, 0.f, 0.f, 0.f, 0.f, 0.f, 0.f, 0.f};
#pragma unroll
        for (int kk = 0; kk < 8; ++kk) {
            v2f a0, a1;
            a0.x = vS[ln * 65 + 8 * kk + 2 * h];
            a0.y = vS[ln * 65 + 8 * kk + 2 * h + 1];
            a1.x = vS[ln * 65 + 8 * kk + 4 + 2 * h];
            a1.y = vS[ln * 65 + 8 * kk + 4 + 2 * h + 1];
            c0 = __builtin_amdgcn_wmma_f32_16x16x4_f32(
                     false, a0, false, eb[2 * kk], (short)0, c0, false, false);
            c1 = __builtin_amdgcn_wmma_f32_16x16x4_f32(
                     false, a1, false, eb[2 * kk + 1], (short)0, c1, false, false);
        }

        // prefetch next timestep's emission tile (global_prefetch_b8 -> GL2;
        // 192MB L2 keeps the 128MB feats tensor resident after first touch)
        if (t + 1 < TT) {
            const float* pf = feats +
                ((size_t)(base + (tid >> 3)) * TT + (t + 1)) * KK + (size_t)(tid & 7) * 8;
            __builtin_prefetch(pf, 0, 3);
        }

        // alpha' = log(D) + rowmax + emit ; write back to alphaS
        const int n = 16 * wv + ln;
#pragma unroll
        for (int vv = 0; vv < 8; ++vv) {
            int m = vv + 8 * h;
            float em = feats[((size_t)(base + m) * TT + t) * KK + n];
            alphaS[m * 65 + n] = __logf(c0[vv] + c1[vv]) + mS[m] + em;
        }
        __syncthreads();
    }

    // terminal logsumexp + gold score -> nll, one thread per batch row
    if (tid < 16) {
        int m = tid;
        int b = base + m;
        const float* tstop = trans + 63 * KK;   // transitions[STOP, n]
        float mx = -3.0e38f;
        for (int nn = 0; nn < KK; ++nn) mx = fmaxf(mx, alphaS[m * 65 + nn] + tstop[nn]);
        float s = 0.0f;
        for (int nn = 0; nn < KK; ++nn) s += __expf(alphaS[m * 65 + nn] + tstop[nn] - mx);
        float fwd = mx + __logf(s);

        float g = 0.0f;
        int prev = 0;  // START
        const int* tg = tags + (size_t)b * TT;
        const float* fb = feats + (size_t)b * TT * KK;
        for (int t = 0; t < TT; ++t) {
            int nt = tg[t];
            g += trans[nt * KK + prev] + fb[(size_t)t * KK + nt];
            prev = nt;
        }
        g += trans[63 * KK + prev];
        nll[b] = fwd - g;
    }
}

// ---------------------------------------------------------------------------
// Viterbi: one wave per batch row, 2 states per lane; transitions transposed
// in LDS with stride 65 (conflict-free); u8 backpointers to workspace.
// ---------------------------------------------------------------------------
__global__ __launch_bounds__(128) void crf_vit(
    const float* __restrict__ feats, const float* __restrict__ trans,
    uint8_t* __restrict__ bp, float* __restrict__ vit_score,
    int* __restrict__ last)
{
    __shared__ float transT[64 * 65];   // transT[p*65+n] = trans[n][p]
    __shared__ float alphaV[4][64];
    __shared__ float termS[4][64];

    const int tid = threadIdx.x, wv = tid >> 5, lane = tid & 31;
    const int b = blockIdx.x * 4 + wv;

    for (int i = tid; i < KK * KK; i += 128) {
        int nn = i >> 6, p = i & 63;
        transT[p * 65 + nn] = trans[i];
    }
    alphaV[wv][lane] = (lane == 0) ? 0.0f : NEGV;
    alphaV[wv][lane + 32] = NEGV;
    __syncthreads();

    uint8_t* bpb = bp + (size_t)b * TT * KK;
    float a0 = NEGV, a1 = NEGV;
    for (int t = 0; t < TT; ++t) {
        float best0 = -3.0e38f, best1 = -3.0e38f;
        int i0 = 0, i1 = 0;
#pragma unroll 8
        for (int p = 0; p < KK; ++p) {
            float ap = alphaV[wv][p];
            float s0 = ap + transT[p * 65 + lane];
            float s1 = ap + transT[p * 65 + lane + 32];
            if (s0 > best0) { best0 = s0; i0 = p; }
            if (s1 > best1) { best1 = s1; i1 = p; }
        }
        const float* fr = feats + ((size_t)b * TT + t) * KK;
        a0 = best0 + fr[lane];
        a1 = best1 + fr[lane + 32];
        bpb[(size_t)t * KK + lane] = (uint8_t)i0;
        bpb[(size_t)t * KK + lane + 32] = (uint8_t)i1;
        alphaV[wv][lane] = a0;
        alphaV[wv][lane + 32] = a1;
        __syncthreads();
    }

    termS[wv][lane] = a0 + trans[63 * KK + lane];
    termS[wv][lane + 32] = a1 + trans[63 * KK + lane + 32];
    __syncthreads();
    if (lane == 0) {
        float bv = -3.0e38f;
        int bi = 0;
        for (int nn = 0; nn < KK; ++nn) {
            float v = termS[wv][nn];
            if (v > bv) { bv = v; bi = nn; }
        }
        vit_score[b] = bv;
        last[b] = bi;
    }
}

// ---------------------------------------------------------------------------
// Backtrace: one thread per batch row; path written as floats.
// ---------------------------------------------------------------------------
__global__ void crf_back(const uint8_t* __restrict__ bp,
                         const int* __restrict__ last,
                         float* __restrict__ path)
{
    int b = blockIdx.x * blockDim.x + threadIdx.x;
    if (b >= BB) return;
    int tag = last[b];
    const uint8_t* bpb = bp + (size_t)b * TT * KK;
    for (int t = TT - 1; t >= 0; --t) {
        path[(size_t)b * TT + t] = (float)tag;
        tag = (int)bpb[(size_t)t * KK + tag];
    }
}

// ---------------------------------------------------------------------------
extern "C" void kernel_launch(void* const* d_in, const int* in_sizes, int n_in,
                              void* d_out, int out_size, void* d_ws, size_t ws_size,
                              hipStream_t stream) {
    const float* feats = (const float*)d_in[0];          // (B,T,K) f32
    const int*   tags  = (const int*)d_in[1];            // (B,T) i32
    const float* trans = (const float*)d_in[2];          // (K,K) f32

    float* out = (float*)d_out;
    float* nll  = out;                 // [0, B)
    float* vsc  = out + BB;            // [B, 2B)
    float* path = out + 2 * BB;        // [2B, 2B + B*T)

    char* ws = (char*)d_ws;
    v2f*     ebuf = (v2f*)ws;                                   // 16 KB
    uint8_t* bp   = (uint8_t*)(ws + 16384);                     // B*T*K = 32 MB
    int*     last = (int*)(ws + 16384 + (size_t)BB * TT * KK);  // B ints

    crf_prep<<<16, 128, 0, stream>>>(trans, ebuf);
    crf_fwd <<<64, 128, 0, stream>>>(feats, tags, trans, ebuf, nll);
    crf_vit <<<BB / 4, 128, 0, stream>>>(feats, trans, bp, vsc, last);
    crf_back<<<BB / 128, 128, 0, stream>>>(bp, last, path);
}